// MHA_52527450030447
// MI455X (gfx1250) — compile-verified
//
#include <hip/hip_runtime.h>
#include <math.h>

typedef __attribute__((ext_vector_type(2))) float v2f;
typedef __attribute__((ext_vector_type(4))) float v4f;
typedef __attribute__((ext_vector_type(8))) float v8f;

#define D_MODEL 1024
#define NHEAD   16
#define HD      64
#define SCTX    8192
#define BATCH   32

// -------------------------------------------------------------------------
// Out[32, ncols] = X[32,1024] @ W[1024,ncols] + bias   via V_WMMA_F32_16X16X4_F32
// One wave (32 threads) per 16-column N tile; both 16-row M tiles per wave so
// each B tile is fetched exactly once.
// ISA layouts (CDNA5 ISA §7.12.2):
//   A 16x4 : lane%16 -> M, lane/16 selects K pair {0,1}|{2,3}, VGPR0=even K, VGPR1=odd K
//   B 4x16 : lane%16 -> N, lane/16 selects K pair,            VGPR0=even K, VGPR1=odd K
//   D 16x16: VGPR r -> M = r + (lane/16)*8, N = lane%16
// -------------------------------------------------------------------------
__global__ __launch_bounds__(32) void proj_wmma_f32(const float* __restrict__ X,
                                                    const float* __restrict__ W,
                                                    const float* __restrict__ bias,
                                                    float* __restrict__ Out,
                                                    int ncols) {
  const int lane = threadIdx.x;
  const int half = lane >> 4;
  const int l16  = lane & 15;
  const size_t n = (size_t)blockIdx.x * 16 + (size_t)l16;

  v8f c0 = {};
  v8f c1 = {};
  const float* xr0 = X + (size_t)l16 * D_MODEL;         // rows 0..15
  const float* xr1 = X + (size_t)(l16 + 16) * D_MODEL;  // rows 16..31

#pragma unroll 4
  for (int k0 = 0; k0 < D_MODEL; k0 += 4) {
    const int ka = k0 + half * 2;                 // this lane's even K
    const float2 a0f = *(const float2*)(xr0 + ka);
    const float2 a1f = *(const float2*)(xr1 + ka);
    v2f a0 = { a0f.x, a0f.y };
    v2f a1 = { a1f.x, a1f.y };
    v2f bb = { W[(size_t)ka * (size_t)ncols + n],
               W[(size_t)(ka + 1) * (size_t)ncols + n] };
    // 8 args: (neg_a, A, neg_b, B, c_mod, C, reuse_a, reuse_b)
    c0 = __builtin_amdgcn_wmma_f32_16x16x4_f32(false, a0, false, bb, (short)0, c0, false, false);
    c1 = __builtin_amdgcn_wmma_f32_16x16x4_f32(false, a1, false, bb, (short)0, c1, false, false);
  }

  const float bv = bias[n];
#pragma unroll
  for (int r = 0; r < 8; ++r) {
    const int m = r + half * 8;
    Out[(size_t)m * (size_t)ncols + n]        = c0[r] + bv;
    Out[(size_t)(m + 16) * (size_t)ncols + n] = c1[r] + bv;
  }
}

// -------------------------------------------------------------------------
// Decode attention: one block per (b,h), 256 threads = 16 groups of 16 lanes.
// Group g processes keys s = g, g+16, ... ; lane l holds float4 slice [4l..4l+3]
// of the 64-dim head. A wave's 32 lanes cover 2 consecutive keys => every K/V
// load instruction touches 512 contiguous bytes (fully coalesced stream).
// Online softmax kept per group, merged across the 16 groups through LDS.
// -------------------------------------------------------------------------
__global__ __launch_bounds__(256) void attn_decode(const float* __restrict__ qkv,   // [32,3072]
                                                   const float* __restrict__ K,     // [32,16,8192,64]
                                                   const float* __restrict__ V,     // [32,16,8192,64]
                                                   float* __restrict__ Y) {         // [32,1024]
  const int bh = blockIdx.x;
  const int b  = bh >> 4;
  const int h  = bh & 15;
  const int tid = threadIdx.x;
  const int g   = tid >> 4;   // key-group 0..15
  const int l   = tid & 15;   // lane within group

  __shared__ __align__(16) float q_s[HD];
  __shared__ float red_m[16];
  __shared__ float red_l[16];
  __shared__ __align__(16) float red_acc[16][HD];

  const float* qp = qkv + (size_t)b * (3 * D_MODEL) + h * HD;
  if (tid < HD) q_s[tid] = qp[tid] * 0.125f;   // fold 1/sqrt(64) into q
  __syncthreads();

  const v4f qq = *(const v4f*)(q_s + l * 4);

  const size_t base4 = (size_t)bh * SCTX * (HD / 4);   // in float4 units
  const v4f* Kp = (const v4f*)K + base4;
  const v4f* Vp = (const v4f*)V + base4;

  float m_run = -INFINITY;
  float l_run = 0.0f;
  v4f   acc   = {0.0f, 0.0f, 0.0f, 0.0f};

  for (int s = g; s < SCTX; s += 16) {
    const v4f kk = __builtin_nontemporal_load(Kp + (size_t)s * 16 + l);
    float d = kk.x * qq.x + kk.y * qq.y + kk.z * qq.z + kk.w * qq.w;
    d += __shfl_xor(d, 1, 32);
    d += __shfl_xor(d, 2, 32);
    d += __shfl_xor(d, 4, 32);
    d += __shfl_xor(d, 8, 32);

    const float m_new = fmaxf(m_run, d);
    const float corr  = __expf(m_run - m_new);
    const float p     = __expf(d - m_new);
    const v4f vv = __builtin_nontemporal_load(Vp + (size_t)s * 16 + l);
    l_run = l_run * corr + p;
    acc   = acc * corr + p * vv;
    m_run = m_new;
  }

  // Current token (position S): group 0 folds it in from the qkv buffer.
  if (g == 0) {
    const float* kc = qkv + (size_t)b * (3 * D_MODEL) + D_MODEL     + h * HD;
    const float* vc = qkv + (size_t)b * (3 * D_MODEL) + 2 * D_MODEL + h * HD;
    const v4f kk = *(const v4f*)(kc + l * 4);
    float d = (kk.x * qq.x + kk.y * qq.y + kk.z * qq.z + kk.w * qq.w);
    d += __shfl_xor(d, 1, 32);
    d += __shfl_xor(d, 2, 32);
    d += __shfl_xor(d, 4, 32);
    d += __shfl_xor(d, 8, 32);
    const float m_new = fmaxf(m_run, d);
    const float corr  = __expf(m_run - m_new);
    const float p     = __expf(d - m_new);
    const v4f vv = *(const v4f*)(vc + l * 4);
    l_run = l_run * corr + p;
    acc   = acc * corr + p * vv;
    m_run = m_new;
  }

  if (l == 0) { red_m[g] = m_run; red_l[g] = l_run; }
  *(v4f*)(&red_acc[g][l * 4]) = acc;
  __syncthreads();

  if (tid < HD) {
    float M = -INFINITY;
#pragma unroll
    for (int i = 0; i < 16; ++i) M = fmaxf(M, red_m[i]);
    float L = 0.0f, num = 0.0f;
#pragma unroll
    for (int i = 0; i < 16; ++i) {
      const float w = __expf(red_m[i] - M);
      L   += red_l[i] * w;
      num += red_acc[i][tid] * w;
    }
    Y[(size_t)b * D_MODEL + h * HD + tid] = num / L;
  }
}

// -------------------------------------------------------------------------
extern "C" void kernel_launch(void* const* d_in, const int* in_sizes, int n_in,
                              void* d_out, int out_size, void* d_ws, size_t ws_size,
                              hipStream_t stream) {
  const float* x  = (const float*)d_in[0];   // [32,1,1024]
  const float* ck = (const float*)d_in[1];   // [32,16,8192,64]
  const float* cv = (const float*)d_in[2];   // [32,16,8192,64]
  const float* Wa = (const float*)d_in[3];   // [1024,3072]
  const float* ba = (const float*)d_in[4];   // [3072]
  const float* Wp = (const float*)d_in[5];   // [1024,1024]
  const float* bp = (const float*)d_in[6];   // [1024]
  float* out = (float*)d_out;                // [32,1,1024]

  float* qkv = (float*)d_ws;                              // 32*3072 floats
  float* y   = qkv + (size_t)BATCH * 3 * D_MODEL;         // 32*1024 floats

  // 1) qkv = x @ W_attn + b_attn      (192 N-tiles, one wave each)
  proj_wmma_f32<<<(3 * D_MODEL) / 16, 32, 0, stream>>>(x, Wa, ba, qkv, 3 * D_MODEL);

  // 2) streaming online-softmax attention over the 2.15 GB KV cache
  attn_decode<<<BATCH * NHEAD, 256, 0, stream>>>(qkv, ck, cv, y);

  // 3) out = y @ W_proj + b_proj      (64 N-tiles)
  proj_wmma_f32<<<D_MODEL / 16, 32, 0, stream>>>(y, Wp, bp, out, D_MODEL);
}